// Gemma3Attention_22333829940034
// MI455X (gfx1250) — compile-verified
//
#include <hip/hip_runtime.h>
#include <hip/hip_bf16.h>

// ---- problem constants (match reference) ----
#define Bc   2
#define Tc   2048
#define Cc   2048
#define Hc   16
#define KVc  4
#define HDc  128
#define NREP 4          // Hc / KVc
#define CAPc 50.0f
#define EPSc 1e-6f
#define SCALEc 0.08838834764831845f   // 1/sqrt(128)

// padded LDS strides (bank-conflict avoidance, keep 8B alignment -> even)
#define QK_STRIDE 130   // 16x128 tiles, row stride 130 words (130%64==2)
#define VT_STRIDE 18    // 128x16 transposed V, row stride 18 words
#define P_STRIDE  18    // 16x16 P tile

typedef __attribute__((ext_vector_type(2))) float v2f;
typedef __attribute__((ext_vector_type(8))) float v8f;

static __device__ __forceinline__ v8f wmma4(v2f a, v2f b, v8f c) {
  // D = A(16x4,f32) * B(4x16,f32) + C(16x16,f32)
  return __builtin_amdgcn_wmma_f32_16x16x4_f32(false, a, false, b, (short)0, c,
                                               false, false);
}

// fragment set for one K-step of the 32x64 wave tile
struct Frag { v2f a0, a1, b0, b1, b2, b3; };

static __device__ __forceinline__ Frag load_frag(const float* __restrict__ ap0,
                                                 const float* __restrict__ ap1,
                                                 const float* __restrict__ bp,
                                                 int N) {
  Frag f;
  f.a0 = *(const v2f*)ap0;                 // A rows lidx / lidx+16, K = ka..ka+1
  f.a1 = *(const v2f*)ap1;
  f.b0.x = bp[0];   f.b0.y = bp[N];        // B cols lidx + {0,16,32,48}
  f.b1.x = bp[16];  f.b1.y = bp[N + 16];
  f.b2.x = bp[32];  f.b2.y = bp[N + 32];
  f.b3.x = bp[48];  f.b3.y = bp[N + 48];
  return f;
}

static __device__ __forceinline__ void do_wmma(const Frag& f, v8f acc[2][4]) {
  acc[0][0] = wmma4(f.a0, f.b0, acc[0][0]);
  acc[0][1] = wmma4(f.a0, f.b1, acc[0][1]);
  acc[0][2] = wmma4(f.a0, f.b2, acc[0][2]);
  acc[0][3] = wmma4(f.a0, f.b3, acc[0][3]);
  acc[1][0] = wmma4(f.a1, f.b0, acc[1][0]);
  acc[1][1] = wmma4(f.a1, f.b1, acc[1][1]);
  acc[1][2] = wmma4(f.a1, f.b2, acc[1][2]);
  acc[1][3] = wmma4(f.a1, f.b3, acc[1][3]);
}

// =====================================================================
// fp32 WMMA GEMM: C[M,N] = A[M,K] @ B[K,N], row-major.
// blockDim = 128 (4 waves, 2x2); wave tile 32x64 (2x4 WMMA accumulators),
// block tile 64x128. M%64==0, N%128==0, K%8==0. Software-pipelined:
// next K-step fragments are in flight while current WMMAs execute.
// =====================================================================
__global__ void gemm_wmma_f32(const float* __restrict__ A,
                              const float* __restrict__ Bm,
                              float* __restrict__ Cm,
                              int M, int N, int Kd) {
  const int lane = threadIdx.x & 31;
  const int wave = threadIdx.x >> 5;           // 0..3
  const int half = lane >> 4;                  // 0 or 1
  const int lidx = lane & 15;
  const int tileM0 = blockIdx.x * 64 + (wave >> 1) * 32;
  const int tileN0 = blockIdx.y * 128 + (wave & 1) * 64;

  v8f acc[2][4] = {};

  const float* ap0 = A + (size_t)(tileM0 + lidx) * Kd + half * 2;
  const float* ap1 = A + (size_t)(tileM0 + 16 + lidx) * Kd + half * 2;
  const float* bp  = Bm + (size_t)(half * 2) * N + tileN0 + lidx;
  const size_t bstep = (size_t)4 * N;

  Frag cur = load_frag(ap0, ap1, bp, N);       // prologue

  for (int k0 = 4; k0 < Kd; k0 += 4) {
    ap0 += 4; ap1 += 4; bp += bstep;
    Frag nxt = load_frag(ap0, ap1, bp, N);     // issue next set first ...

    // speculative stream-ahead (silently dropped if OOB per ISA)
    __builtin_prefetch(ap0 + 60, 0, 1);
    __builtin_prefetch(bp + (size_t)12 * N, 0, 1);

    do_wmma(cur, acc);                         // ... then crunch current set
    cur = nxt;
  }
  do_wmma(cur, acc);                           // epilogue

  // C/D layout: component v of half `half` is row (half*8 + v), col lidx.
  for (int i = 0; i < 2; ++i)
    for (int j = 0; j < 4; ++j)
      for (int v = 0; v < 8; ++v)
        Cm[(size_t)(tileM0 + i * 16 + half * 8 + v) * N + tileN0 + j * 16 + lidx] =
            acc[i][j][v];
}

// =====================================================================
// Fused RMSNorm + RoPE, in place. One wave per (b,t,head) vector of 128.
// hbuf layout: [B*T, nheads, HD] contiguous.
// =====================================================================
__global__ void qk_norm_rope_kernel(float* __restrict__ hbuf,
                                    const float* __restrict__ w,
                                    const float* __restrict__ fc,
                                    const float* __restrict__ fs,
                                    int nheads) {
  const int lane = threadIdx.x & 31;
  const int wid  = blockIdx.x * (blockDim.x >> 5) + (threadIdx.x >> 5);
  const int bt   = wid / nheads;               // wid = bt*nheads + h
  const int t    = bt & (Tc - 1);

  float* base = hbuf + (size_t)wid * HDc;
  float4 e = ((const float4*)base)[lane];      // dims lane*4 .. lane*4+3

  float ss = e.x * e.x + e.y * e.y + e.z * e.z + e.w * e.w;
  for (int m = 16; m >= 1; m >>= 1) ss += __shfl_xor(ss, m, 32);
  const float r = rsqrtf(ss * (1.0f / HDc) + EPSc);

  float4 wv = ((const float4*)w)[lane];
  const float re0 = e.x * r * wv.x, im0 = e.y * r * wv.y;
  const float re1 = e.z * r * wv.z, im1 = e.w * r * wv.w;

  const int j0 = lane * 2;                     // rope pair indices j0, j0+1
  const float c0 = fc[t * (HDc / 2) + j0],     s0 = fs[t * (HDc / 2) + j0];
  const float c1 = fc[t * (HDc / 2) + j0 + 1], s1 = fs[t * (HDc / 2) + j0 + 1];

  float4 o;
  o.x = re0 * c0 - im0 * s0;  o.y = re0 * s0 + im0 * c0;
  o.z = re1 * c1 - im1 * s1;  o.w = re1 * s1 + im1 * c1;
  ((float4*)base)[lane] = o;
}

// =====================================================================
// Flash attention, fp32 WMMA. 4 waves/block own 4 consecutive q-tiles of
// the same (b,h); the block cooperatively stages each K tile and a
// transposed V tile into LDS once per k-iteration (shared across waves).
// Q: [B*T, H, HD]  K,V: [B*T, KV, HD]  Y: [B*T, H, HD]
// =====================================================================
__global__ void attn_kernel(const float* __restrict__ Qm,
                            const float* __restrict__ Km,
                            const float* __restrict__ Vm,
                            float* __restrict__ Ym) {
  __shared__ float ldsQ[4][16 * QK_STRIDE];    // ~33 KB
  __shared__ float ldsK[16 * QK_STRIDE];       // ~8 KB
  __shared__ float ldsVt[HDc * VT_STRIDE];     // ~9 KB (V transposed: [d][k])
  __shared__ float ldsP[4][16 * P_STRIDE];     // ~4.5 KB (C/D -> A relay)

  const int tid  = threadIdx.x;
  const int lane = tid & 31;
  const int wave = tid >> 5;
  const int half = lane >> 4;
  const int lidx = lane & 15;

  const int bh   = blockIdx.x >> 5;            // 32 blocks per (b,h)
  const int qt0  = (blockIdx.x & 31) * 4;      // first q-tile of this block
  const int myqt = qt0 + wave;
  const int h    = bh & (Hc - 1);
  const int b    = bh >> 4;
  const int kvh  = h >> 2;                     // GQA: h / NREP
  const int qbase = myqt * 16;

  // stage this wave's Q tile (16 x 128) into its LDS slab (coalesced)
  const float* qsrc = Qm + ((size_t)(b * Tc + qbase) * Hc + h) * HDc;
  for (int idx = lane; idx < 16 * HDc; idx += 32) {
    const int rr = idx >> 7, dd = idx & 127;
    ldsQ[wave][rr * QK_STRIDE + dd] = qsrc[(size_t)rr * Hc * HDc + dd];
  }

  v8f acc[8] = {};                             // O tile: 16 x 128
  float m_i[8], l_i[8];
  for (int v = 0; v < 8; ++v) { m_i[v] = -1e30f; l_i[v] = 0.0f; }

  const int ktmax = qt0 + 3;                   // block-uniform trip count
  for (int kt = 0; kt <= ktmax; ++kt) {
    const int kbase = kt * 16;

    __syncthreads();                           // previous tile fully consumed
    // cooperative stage: K tile row-major, V tile transposed ([d][k])
    {
      const float* kg = Km + ((size_t)(b * Tc + kbase) * KVc + kvh) * HDc;
      const float* vg = Vm + ((size_t)(b * Tc + kbase) * KVc + kvh) * HDc;
      for (int idx = tid; idx < 16 * HDc; idx += 128) {
        const int r = idx >> 7, d = idx & 127;
        const size_t go = (size_t)r * (KVc * HDc) + d;
        ldsK[r * QK_STRIDE + d] = kg[go];
        ldsVt[d * VT_STRIDE + r] = vg[go];
      }
    }
    __syncthreads();

    if (kt <= myqt) {                          // wave-uniform predicate
      const bool diag = (kt == myqt);

      // ---- S = Q @ K^T (16x16), K-loop over HD in steps of 4 ----
      v8f s = {};
      for (int ks = 0; ks < HDc / 4; ++ks) {
        const int kk = ks * 4 + half * 2;
        v2f a  = *(const v2f*)&ldsQ[wave][lidx * QK_STRIDE + kk];
        v2f bf = *(const v2f*)&ldsK[lidx * QK_STRIDE + kk];
        s = wmma4(a, bf, s);
      }

      // ---- softcap + causal mask + online softmax ----
      float mrow[8];
      for (int v = 0; v < 8; ++v) {
        float sv = CAPc * tanhf(s[v] * (SCALEc / CAPc));
        const int rowg = qbase + half * 8 + v;
        sv = (diag && (kbase + lidx > rowg)) ? -1e30f : sv;
        s[v] = sv;
        float x = sv;
        x = fmaxf(x, __shfl_xor(x, 1, 32));
        x = fmaxf(x, __shfl_xor(x, 2, 32));
        x = fmaxf(x, __shfl_xor(x, 4, 32));
        x = fmaxf(x, __shfl_xor(x, 8, 32));
        mrow[v] = x;
      }
      for (int v = 0; v < 8; ++v) {
        const float mn    = fmaxf(m_i[v], mrow[v]);
        const float alpha = __expf(m_i[v] - mn);
        m_i[v] = mn;
        const float pv = __expf(s[v] - mn);
        float rs = pv;
        rs += __shfl_xor(rs, 1, 32);
        rs += __shfl_xor(rs, 2, 32);
        rs += __shfl_xor(rs, 4, 32);
        rs += __shfl_xor(rs, 8, 32);
        l_i[v] = l_i[v] * alpha + rs;
        for (int j = 0; j < 8; ++j) acc[j][v] *= alpha;
        ldsP[wave][(v + half * 8) * P_STRIDE + lidx] = pv;  // C/D -> A relay
      }

      // ---- O += P @ V (A frag from own LDS slab; same-wave DS ordering) ----
      for (int ks = 0; ks < 4; ++ks) {
        const int kk = ks * 4 + half * 2;
        v2f pa = *(const v2f*)&ldsP[wave][lidx * P_STRIDE + kk];
        for (int j = 0; j < 8; ++j) {
          v2f bv = *(const v2f*)&ldsVt[(j * 16 + lidx) * VT_STRIDE + kk];
          acc[j] = wmma4(pa, bv, acc[j]);
        }
      }
    }
  }

  // ---- normalize + write Y[b, t, h*HD + d] ----
  for (int v = 0; v < 8; ++v) {
    const float inv = 1.0f / l_i[v];
    float* yo = Ym + ((size_t)(b * Tc + qbase + half * 8 + v) * Hc + h) * HDc;
    for (int j = 0; j < 8; ++j) yo[j * 16 + lidx] = acc[j][v] * inv;
  }
}

// =====================================================================
// launch
// =====================================================================
extern "C" void kernel_launch(void* const* d_in, const int* in_sizes, int n_in,
                              void* d_out, int out_size, void* d_ws, size_t ws_size,
                              hipStream_t stream) {
  const float* x   = (const float*)d_in[0];
  const float* Wq  = (const float*)d_in[1];
  const float* Wk  = (const float*)d_in[2];
  const float* Wv  = (const float*)d_in[3];
  const float* Wo  = (const float*)d_in[4];
  const float* qnw = (const float*)d_in[5];
  const float* knw = (const float*)d_in[6];
  const float* fc  = (const float*)d_in[7];
  const float* fs  = (const float*)d_in[8];
  // d_in[9] (mask) handled analytically (causal)

  float* ws = (float*)d_ws;
  float* Q  = ws;                                   // 8388608 floats
  float* Kx = Q  + (size_t)Bc * Tc * Hc * HDc;      // 2097152
  float* Vx = Kx + (size_t)Bc * Tc * KVc * HDc;     // 2097152
  float* Y  = Vx + (size_t)Bc * Tc * KVc * HDc;     // 8388608
  float* out = (float*)d_out;

  const int M = Bc * Tc;                            // 4096
  dim3 blk(128);

  // 1) projections (N multiples of 128: 2048 and 512)
  gemm_wmma_f32<<<dim3(M / 64, (Hc * HDc) / 128), blk, 0, stream>>>(x, Wq, Q,  M, Hc * HDc,  Cc);
  gemm_wmma_f32<<<dim3(M / 64, (KVc * HDc) / 128), blk, 0, stream>>>(x, Wk, Kx, M, KVc * HDc, Cc);
  gemm_wmma_f32<<<dim3(M / 64, (KVc * HDc) / 128), blk, 0, stream>>>(x, Wv, Vx, M, KVc * HDc, Cc);

  // 2) RMSNorm + RoPE on Q and K (8 waves / 256-thread block, exact grids)
  qk_norm_rope_kernel<<<(M * Hc) / 8,  256, 0, stream>>>(Q,  qnw, fc, fs, Hc);
  qk_norm_rope_kernel<<<(M * KVc) / 8, 256, 0, stream>>>(Kx, knw, fc, fs, KVc);

  // 3) flash attention: B*H blocks x 32, each block = 4 q-tiles
  attn_kernel<<<Bc * Hc * 32, 128, 0, stream>>>(Q, Kx, Vx, Y);

  // 4) output projection
  gemm_wmma_f32<<<dim3(M / 64, Cc / 128), blk, 0, stream>>>(Y, Wo, out, M, Cc, Cc);
}